// SparseMatrixModule_18098992185813
// MI455X (gfx1250) — compile-verified
//
#include <hip/hip_runtime.h>
#include <stdint.h>

#define WAVES_PER_BLOCK 8
#define BLOCK_THREADS   (WAVES_PER_BLOCK * 32)
#define CHUNK           32   // nonzeros staged per wave per chunk

// ---------------------------------------------------------------------------
// Zero the output (harness poisons d_out with 0xAA before timing).
// ---------------------------------------------------------------------------
__global__ void spmm_zero_f32(float* __restrict__ p, int n) {
  long long i  = (long long)blockIdx.x * blockDim.x + threadIdx.x;
  long long i4 = i * 4;
  if (i4 + 4 <= n) {
    *(float4*)(p + i4) = make_float4(0.f, 0.f, 0.f, 0.f);
  } else if (i4 < n) {
    for (long long k = i4; k < n; ++k) p[k] = 0.f;
  }
}

// ---------------------------------------------------------------------------
// Issue one chunk of 32 nonzeros' (row,col,value) as async global->LDS copies.
// GVS mode: mem_addr = SGPR_base + VGPR_byte_offset. 3 async instrs per wave,
// tracked by ASYNCcnt; completion is in issue order.
// ---------------------------------------------------------------------------
__device__ __forceinline__ void issue_chunk_async(
    const long long* __restrict__ row,
    const long long* __restrict__ col,
    const float*     __restrict__ value,
    long long nnz, long long base, int lane,
    uint32_t lrow, uint32_t lcol, uint32_t lval)
{
  long long gi = base + lane;
  if (gi >= nnz) gi = nnz - 1;                 // clamp: keep loads in-bounds
  uint32_t o8 = (uint32_t)(gi << 3);           // byte offset into int64 arrays
  uint32_t o4 = (uint32_t)(gi << 2);           // byte offset into float array
  asm volatile("global_load_async_to_lds_b64 %0, %1, %2"
               :: "v"(lrow), "v"(o8), "s"((uint64_t)(uintptr_t)row) : "memory");
  asm volatile("global_load_async_to_lds_b64 %0, %1, %2"
               :: "v"(lcol), "v"(o8), "s"((uint64_t)(uintptr_t)col) : "memory");
  asm volatile("global_load_async_to_lds_b32 %0, %1, %2"
               :: "v"(lval), "v"(o4), "s"((uint64_t)(uintptr_t)value) : "memory");
}

// ---------------------------------------------------------------------------
// Process 16 nonzeros (locals h*16 .. h*16+15, two concurrent per wave step)
// in three phases so the 8 b128 gathers issue back-to-back and overlap:
//   (1) LDS index reads  (2) x-row gathers  (3) y scatter atomics
// All global addressing uses 32-bit byte offsets (x/y are 25.6 MB).
// ---------------------------------------------------------------------------
__device__ __forceinline__ void process_half(
    const float* __restrict__ x, float* __restrict__ y,
    const long long* rowbuf, const long long* colbuf, const float* valbuf,
    int sub, uint32_t fbytes)
{
  uint32_t xoff[8], yoff[8];
  float    vals[8];
#pragma unroll
  for (int j = 0; j < 8; ++j) {
    const int local = j * 2 + sub;                       // 0..15
    const uint32_t cc = (uint32_t)colbuf[local];
    const uint32_t rr = (uint32_t)rowbuf[local];
    vals[j] = valbuf[local];
    xoff[j] = cc * 256u + fbytes;                        // D*4 = 256 B per row
    yoff[j] = rr * 256u + fbytes;
  }
  float4 xv[8];
#pragma unroll
  for (int j = 0; j < 8; ++j)
    xv[j] = *(const float4*)((const char*)x + xoff[j]);  // L2-resident gather
#pragma unroll
  for (int j = 0; j < 8; ++j) {
    float* yp = (float*)((char*)y + yoff[j]);
    const float v = vals[j];
    unsafeAtomicAdd(yp + 0, xv[j].x * v);                // non-returning f32
    unsafeAtomicAdd(yp + 1, xv[j].y * v);                // atomics (no RET)
    unsafeAtomicAdd(yp + 2, xv[j].z * v);
    unsafeAtomicAdd(yp + 3, xv[j].w * v);
  }
}

// ---------------------------------------------------------------------------
// COO SpMM scatter:  y[row[i], :] += value[i] * x[col[i], :]   (D = 64 f32)
// One wave handles 2 nonzeros per step (16 lanes x float4 = 64 features).
// Index stream double-buffered in LDS via async copies.
// ---------------------------------------------------------------------------
__global__ void __launch_bounds__(BLOCK_THREADS)
spmm_coo_scatter(const float*     __restrict__ x,
                 const long long* __restrict__ row,
                 const long long* __restrict__ col,
                 const float*     __restrict__ value,
                 float*           __restrict__ y,
                 long long nnz)
{
  __shared__ long long s_row[2][WAVES_PER_BLOCK][CHUNK];
  __shared__ long long s_col[2][WAVES_PER_BLOCK][CHUNK];
  __shared__ float     s_val[2][WAVES_PER_BLOCK][CHUNK];

  const int lane = threadIdx.x & 31;
  const int wave = threadIdx.x >> 5;

  const long long wave_id   = (long long)blockIdx.x * WAVES_PER_BLOCK + wave;
  const long long num_waves = (long long)gridDim.x * WAVES_PER_BLOCK;

  const long long total_chunks = (nnz + CHUNK - 1) / CHUNK;
  const long long per  = (total_chunks + num_waves - 1) / num_waves;
  const long long c0   = wave_id * per;
  long long cend = c0 + per;
  if (cend > total_chunks) cend = total_chunks;
  if (c0 >= cend) return;

  // Per-lane LDS byte offsets (low 32 bits of the generic pointer = LDS offset)
  uint32_t lrow[2], lcol[2], lval[2];
#pragma unroll
  for (int b = 0; b < 2; ++b) {
    lrow[b] = (uint32_t)(uintptr_t)&s_row[b][wave][lane];
    lcol[b] = (uint32_t)(uintptr_t)&s_col[b][wave][lane];
    lval[b] = (uint32_t)(uintptr_t)&s_val[b][wave][lane];
  }

  // Prologue: stage first chunk
  issue_chunk_async(row, col, value, nnz, c0 * CHUNK, lane,
                    lrow[0], lcol[0], lval[0]);

  const int      sub    = lane >> 4;                 // which of 2 nonzeros
  const uint32_t fbytes = (uint32_t)(lane & 15) * 16u;  // feature byte offset

  for (long long c = c0; c < cend; ++c) {
    const int buf = (int)((c - c0) & 1);

    if (c + 1 < cend) {
      // Stage next chunk into the other buffer, then wait for current chunk:
      // the 3 youngest async ops belong to chunk c+1; completion is in order.
      issue_chunk_async(row, col, value, nnz, (c + 1) * CHUNK, lane,
                        lrow[buf ^ 1], lcol[buf ^ 1], lval[buf ^ 1]);
      asm volatile("s_wait_asynccnt 0x3" ::: "memory");
    } else {
      asm volatile("s_wait_asynccnt 0x0" ::: "memory");
    }

    const long long base = c * CHUNK;
    const long long* rowbuf = s_row[buf][wave];
    const long long* colbuf = s_col[buf][wave];
    const float*     valbuf = s_val[buf][wave];

    if (base + CHUNK <= nnz) {
      // Fast path: whole chunk in bounds, no per-nonzero guard.
      process_half(x, y, rowbuf,      colbuf,      valbuf,      sub, fbytes);
      process_half(x, y, rowbuf + 16, colbuf + 16, valbuf + 16, sub, fbytes);
    } else {
      // Rare tail path: guard each nonzero.
      const int rem = (int)(nnz - base);
#pragma unroll
      for (int j = 0; j < CHUNK / 2; ++j) {
        const int local = j * 2 + sub;
        if (local >= rem) continue;
        const uint32_t cc = (uint32_t)colbuf[local];
        const uint32_t rr = (uint32_t)rowbuf[local];
        const float    v  = valbuf[local];
        const float4 xv = *(const float4*)((const char*)x + cc * 256u + fbytes);
        float* yp = (float*)((char*)y + rr * 256u + fbytes);
        unsafeAtomicAdd(yp + 0, xv.x * v);
        unsafeAtomicAdd(yp + 1, xv.y * v);
        unsafeAtomicAdd(yp + 2, xv.z * v);
        unsafeAtomicAdd(yp + 3, xv.w * v);
      }
    }
  }
}

// ---------------------------------------------------------------------------
// Harness entry point.
// Inputs (setup_inputs order): x[f32 N*64], row[i64 NNZ], col[i64 NNZ],
//                              value[f32 NNZ], size[scalar]
// Output: y[f32 N*64]
// ---------------------------------------------------------------------------
extern "C" void kernel_launch(void* const* d_in, const int* in_sizes, int n_in,
                              void* d_out, int out_size, void* d_ws, size_t ws_size,
                              hipStream_t stream) {
  const float*     x     = (const float*)d_in[0];
  const long long* row   = (const long long*)d_in[1];
  const long long* col   = (const long long*)d_in[2];
  const float*     value = (const float*)d_in[3];
  float*           y     = (float*)d_out;
  const long long  nnz   = (long long)in_sizes[1];

  // 1) zero the output
  {
    int n4   = (out_size + 3) / 4;
    int grid = (n4 + 255) / 256;
    spmm_zero_f32<<<grid, 256, 0, stream>>>(y, out_size);
  }

  // 2) scatter-add. 1250 blocks * 8 waves = 10000 waves; with NNZ=1.6M that is
  //    exactly 5 chunks of 32 nonzeros per wave (guards handle other sizes).
  {
    const long long total_chunks = (nnz + CHUNK - 1) / CHUNK;
    long long grid_ll = (total_chunks + (long long)WAVES_PER_BLOCK * 5 - 1) /
                        ((long long)WAVES_PER_BLOCK * 5);
    if (grid_ll > 2048) grid_ll = 2048;
    if (grid_ll < 1)    grid_ll = 1;
    spmm_coo_scatter<<<(int)grid_ll, BLOCK_THREADS, 0, stream>>>(
        x, row, col, value, y, nnz);
  }
}